// StochasticRepopulateFast_7722351198459
// MI455X (gfx1250) — compile-verified
//
#include <hip/hip_runtime.h>
#include <stdint.h>

// ---------------------------------------------------------------------------
// StochasticRepopulate on MI455X (gfx1250)
//
// 100 sequential spread steps on a 2048x2048 f32 raster. Working set (~50MB)
// is fully L2-resident (192MB L2), so each step is an L2-bandwidth-bound
// stencil pass. Tiles (18x66 incl. halo) are DMA'd into LDS by the Tensor
// Data Mover (TENSOR_LOAD_TO_LDS, tracked with TENSORcnt) while the 8 waves
// of the block compute JAX-exact Threefry-2x32 random factors; a
// s_wait_tensorcnt(0) + workgroup barrier then publishes the tile and the
// 3x3 max-pool runs out of LDS. Each workgroup handles a top-half tile and
// its row+1024 partner tile so one Threefry block serves two pixels (JAX
// pairs count i with i+HW/2). Zero-padded ping-pong buffers make every TDM
// tile in-bounds; zero padding == SAME/-inf padding since all values >= 0.
// ---------------------------------------------------------------------------

#define Hh 2048
#define Ww 2048
#define HALF_CNT 2097152u      // H*W/2, threefry pair offset
#define TX 64
#define TY 16
#define LW 66                  // TX+2
#define LH 18                  // TY+2
#define NSLOT (LH*LW)          // 1188
#define PST 2052               // padded row stride (elements), 16B-aligned rows
#define PRR 2050               // padded rows
#define NSPREADS 100

// ---------------- Threefry-2x32, exactly as jax._src.prng ------------------
__host__ __device__ static inline unsigned rotl32(unsigned x, int r) {
  return (x << r) | (x >> (32 - r));
}

__host__ __device__ static inline void threefry2x32(unsigned k0, unsigned k1,
                                                    unsigned c0, unsigned c1,
                                                    unsigned &o0, unsigned &o1) {
  unsigned k2 = k0 ^ k1 ^ 0x1BD11BDAu;
  unsigned x0 = c0 + k0, x1 = c1 + k1;
#define TF_ROUND(rr_) { x0 += x1; x1 = rotl32(x1, rr_); x1 ^= x0; }
  TF_ROUND(13) TF_ROUND(15) TF_ROUND(26) TF_ROUND(6)
  x0 += k1; x1 += k2 + 1u;
  TF_ROUND(17) TF_ROUND(29) TF_ROUND(16) TF_ROUND(24)
  x0 += k2; x1 += k0 + 2u;
  TF_ROUND(13) TF_ROUND(15) TF_ROUND(26) TF_ROUND(6)
  x0 += k0; x1 += k1 + 3u;
  TF_ROUND(17) TF_ROUND(29) TF_ROUND(16) TF_ROUND(24)
  x0 += k1; x1 += k2 + 4u;
  TF_ROUND(13) TF_ROUND(15) TF_ROUND(26) TF_ROUND(6)
  x0 += k2; x1 += k0 + 5u;
#undef TF_ROUND
  o0 = x0; o1 = x1;
}

__device__ static inline float u01(unsigned bits) {
  // jax.random.uniform: bitcast((bits>>9)|0x3f800000) - 1.0  in [0,1)
  return __uint_as_float((bits >> 9) | 0x3F800000u) - 1.0f;
}

// ---------------- Tensor Data Mover (gfx1250) ------------------------------
#if defined(__AMDGCN__) && __has_builtin(__builtin_amdgcn_tensor_load_to_lds) && \
    __has_builtin(__builtin_amdgcn_s_wait_tensorcnt)
#define USE_TDM 1
#else
#define USE_TDM 0
#endif

#if USE_TDM
typedef unsigned int u32x4 __attribute__((ext_vector_type(4)));
typedef int          i32x4 __attribute__((ext_vector_type(4)));
typedef int          i32x8 __attribute__((ext_vector_type(8)));

__device__ static inline unsigned lds_byte_off(const void* p) {
  // Flat shared address = {LDS aperture, 32-bit LDS offset}; low 32 bits are
  // the LDS byte address the TDM descriptor wants (ISA 10.2 aperture calc).
  return (unsigned)(unsigned long long)p;
}

// 2-D tile load: tile LH x LW f32 elements, row stride PST elements.
__device__ static inline void tdm_load_tile(const float* gptr, unsigned lds_byte,
                                            unsigned rem_w, unsigned rem_h) {
  unsigned long long ga = (unsigned long long)gptr;
  u32x4 g0;
  g0[0] = 1u;                                            // count=1 (valid), no gather
  g0[1] = lds_byte;                                      // lds_addr
  g0[2] = (unsigned)ga;                                  // global_addr[31:0]
  g0[3] = (unsigned)((ga >> 32) & 0x1FFFFFFull) | (2u << 30); // addr[56:32] | type=2
  i32x8 g1;
  g1[0] = (int)(2u << 16);                               // data_size = 4B
  g1[1] = (int)((rem_w & 0xFFFFu) << 16);                // tensor_dim0 lo16 -> [31:16]
  g1[2] = (int)((rem_w >> 16) | ((rem_h & 0xFFFFu) << 16)); // td0 hi16 | td1 lo16
  g1[3] = (int)((rem_h >> 16) | ((unsigned)LW << 16));   // td1 hi16 | tile_dim0=66
  g1[4] = (int)LH;                                       // tile_dim1=18, tile_dim2=0
  g1[5] = (int)PST;                                      // tensor_dim0_stride lo32
  g1[6] = 0;                                             // stride hi | dim1_stride (2D: unused)
  g1[7] = 0;
  i32x4 z4 = (i32x4)0;
#if __clang_major__ >= 23
  i32x8 z8 = (i32x8)0;
  __builtin_amdgcn_tensor_load_to_lds(g0, g1, z4, z4, z8, 0);
#else
  __builtin_amdgcn_tensor_load_to_lds(g0, g1, z4, z4, 0);
#endif
}
#endif // USE_TDM

// ---------------- Kernels --------------------------------------------------

// Init: goodness = max(habitat, terrain); xA = zero-padded seed*habitat;
// zero borders of both ping-pong buffers (re-done every call: deterministic).
__global__ __launch_bounds__(256)
void srp_init_kernel(const float* __restrict__ seed, const float* __restrict__ hab,
                     const float* __restrict__ ter, float* __restrict__ xA,
                     float* __restrict__ xB, float* __restrict__ good) {
  size_t i = (size_t)blockIdx.x * 256 + threadIdx.x;
  if (i >= (size_t)PRR * PST) return;
  int prow = (int)(i / PST), pcol = (int)(i % PST);
  int r = prow - 1, c = pcol - 1;
  if (r >= 0 && r < Hh && c >= 0 && c < Ww) {
    int ii = r * Ww + c;
    float h = hab[ii];
    xA[i] = seed[ii] * h;
    good[ii] = fmaxf(h, ter[ii]);
  } else {
    xA[i] = 0.0f;
    xB[i] = 0.0f;
  }
}

// One spread step. Block = 256 threads (8 wave32), processes a 16x64 tile in
// the top half AND its partner tile 1024 rows below (shared Threefry blocks).
__global__ __launch_bounds__(256)
void srp_spread_kernel(const float* __restrict__ xin, float* __restrict__ xout,
                       const float* __restrict__ good, unsigned k0, unsigned k1) {
  __shared__ float s_raw_t[NSLOT];   // raw x, top tile (incl. halo)
  __shared__ float s_raw_b[NSLOT];   // raw x, bottom tile
  __shared__ float s_xs_t[NSLOT];    // randomized xs
  __shared__ float s_xs_b[NSLOT];

  const int tid = threadIdx.x;
  const int bx = blockIdx.x;         // 0..31
  const int by = blockIdx.y;         // 0..63
  const int pc   = bx * TX;          // padded col of halo start
  const int pr_t = by * TY;          // padded row of halo start (top tile)
  const int pr_b = by * TY + (Hh / 2);

#if USE_TDM
  if (tid == 0) {
    tdm_load_tile(xin + (size_t)pr_t * PST + pc, lds_byte_off(s_raw_t),
                  (unsigned)(PST - pc), (unsigned)(PRR - pr_t));
    tdm_load_tile(xin + (size_t)pr_b * PST + pc, lds_byte_off(s_raw_b),
                  (unsigned)(PST - pc), (unsigned)(PRR - pr_b));
  }
#endif

  // Threefry factors for every halo slot — overlaps with the TDM DMA.
  // Slot (ly,lx) -> interior coords rr = by*TY-1+ly (may be -1..1024),
  // cc = bx*TX-1+lx. Pixel i=rr*W+cc pairs with i+HW/2 (row rr+1024):
  // lane0 -> top pixel, lane1 -> bottom pixel. Edge rows at the half seam
  // (rr==-1, rr==1024) pick the correct lane; border slots hold 0 anyway.
  for (int p = 0; p < 5; ++p) {
    int slot = tid + p * 256;
    if (slot < NSLOT) {
      int ly = slot / LW, lx = slot - ly * LW;
      int rr = by * TY - 1 + ly;
      int cc = bx * TX - 1 + lx;
      int jrow = (rr < 0) ? (Hh / 2 - 1) : ((rr > Hh / 2 - 1) ? 0 : rr);
      unsigned j = (unsigned)(jrow * Ww + cc);     // cc=-1/2048: junk, hits border=0
      unsigned o0, o1;
      threefry2x32(k0, k1, j, j + HALF_CNT, o0, o1);
      float ft = 0.9f + 0.1f * u01((rr > Hh / 2 - 1) ? o1 : o0);
      float fb = 0.9f + 0.1f * u01((rr < 0) ? o0 : o1);
#if USE_TDM
      s_xs_t[slot] = ft;
      s_xs_b[slot] = fb;
#else
      float rt = xin[(size_t)(pr_t + ly) * PST + pc + lx];
      float rb = xin[(size_t)(pr_b + ly) * PST + pc + lx];
      s_raw_t[slot] = rt;  s_raw_b[slot] = rb;
      s_xs_t[slot] = rt * ft;
      s_xs_b[slot] = rb * fb;
#endif
    }
  }

#if USE_TDM
  if (tid == 0) __builtin_amdgcn_s_wait_tensorcnt(0);  // issuing wave drains TDM
  __syncthreads();                                     // publish LDS to all waves
  for (int p = 0; p < 5; ++p) {
    int slot = tid + p * 256;
    if (slot < NSLOT) {
      s_xs_t[slot] *= s_raw_t[slot];
      s_xs_b[slot] *= s_raw_b[slot];
    }
  }
#endif
  __syncthreads();

  // 3x3 max-pool from LDS; 4 pixels per tile per thread.
  for (int q = 0; q < 4; ++q) {
    int idx = tid + q * 256;          // 0..1023 within 16x64 tile
    int iy = idx >> 6, ix = idx & 63;
    int ir = by * TY + iy, ic = bx * TX + ix;
    { // top tile
      const float* w = &s_xs_t[iy * LW + ix];
      float m = w[0];
      m = fmaxf(m, w[1]);        m = fmaxf(m, w[2]);
      m = fmaxf(m, w[LW]);       m = fmaxf(m, w[LW + 1]);   m = fmaxf(m, w[LW + 2]);
      m = fmaxf(m, w[2 * LW]);   m = fmaxf(m, w[2 * LW + 1]); m = fmaxf(m, w[2 * LW + 2]);
      float res = fmaxf(m * good[ir * Ww + ic], s_raw_t[(iy + 1) * LW + ix + 1]);
      xout[(size_t)(ir + 1) * PST + (ic + 1)] = res;
    }
    { // bottom tile (row + 1024)
      const float* w = &s_xs_b[iy * LW + ix];
      float m = w[0];
      m = fmaxf(m, w[1]);        m = fmaxf(m, w[2]);
      m = fmaxf(m, w[LW]);       m = fmaxf(m, w[LW + 1]);   m = fmaxf(m, w[LW + 2]);
      m = fmaxf(m, w[2 * LW]);   m = fmaxf(m, w[2 * LW + 1]); m = fmaxf(m, w[2 * LW + 2]);
      int br = ir + Hh / 2;
      float res = fmaxf(m * good[br * Ww + ic], s_raw_b[(iy + 1) * LW + ix + 1]);
      xout[(size_t)(br + 1) * PST + (ic + 1)] = res;
    }
  }
}

// Final: out = x * habitat (un-pad)
__global__ __launch_bounds__(256)
void srp_final_kernel(const float* __restrict__ xfin, const float* __restrict__ hab,
                      float* __restrict__ out) {
  int i = blockIdx.x * 256 + threadIdx.x;
  if (i >= Hh * Ww) return;
  int r = i >> 11, c = i & (Ww - 1);
  out[i] = xfin[(size_t)(r + 1) * PST + (c + 1)] * hab[i];
}

// ---------------- Launch ---------------------------------------------------
extern "C" void kernel_launch(void* const* d_in, const int* in_sizes, int n_in,
                              void* d_out, int out_size, void* d_ws, size_t ws_size,
                              hipStream_t stream) {
  (void)in_sizes; (void)n_in; (void)out_size; (void)ws_size;
  const float* seed    = (const float*)d_in[0];
  const float* habitat = (const float*)d_in[1];
  const float* terrain = (const float*)d_in[2];
  float* out = (float*)d_out;

  const size_t PAD_ELEMS = (size_t)PRR * PST;            // 4,206,600
  float* xA   = (float*)d_ws;                            // padded ping
  float* xB   = xA + PAD_ELEMS;                          // padded pong
  float* good = xB + PAD_ELEMS;                          // 2048^2 goodness
  // ws needed: (2*4,206,600 + 4,194,304)*4 ≈ 48.1 MiB

  // Host-side jax.random.split(jax.random.key(1), 100): key(1) = (0,1).
  unsigned kb[2 * NSPREADS];
  for (int p = 0; p < NSPREADS; ++p)
    threefry2x32(0u, 1u, (unsigned)p, (unsigned)(p + NSPREADS), kb[p], kb[p + NSPREADS]);
  // keys[s] = (out[2s], out[2s+1]) after reshape(100, 2)

  const int initN = (int)PAD_ELEMS;
  srp_init_kernel<<<(initN + 255) / 256, 256, 0, stream>>>(seed, habitat, terrain,
                                                           xA, xB, good);

  dim3 grid(Ww / TX, (Hh / 2) / TY);                     // 32 x 64 blocks
  for (int s = 0; s < NSPREADS; ++s) {
    const float* in = (s & 1) ? xB : xA;
    float*       o  = (s & 1) ? xA : xB;
    srp_spread_kernel<<<grid, 256, 0, stream>>>(in, o, good, kb[2 * s], kb[2 * s + 1]);
  }
  // 100 iterations: last write (s=99, odd) lands in xA.
  srp_final_kernel<<<(Hh * Ww + 255) / 256, 256, 0, stream>>>(xA, habitat, out);
}